// CTCLoss_41515153883164
// MI455X (gfx1250) — compile-verified
//
#include <hip/hip_runtime.h>
#include <hip/hip_bf16.h>
#include <stdint.h>

// CTC loss for B=64, T=2048, C=96, S=256 on gfx1250 (wave32).
// Phase 1: per-(b,t) log-sum-exp normalizers (bandwidth-trivial).
// Phase 2: one workgroup per batch element runs the 2048-step trellis with
//          one barrier per step; next row is prefetched into LDS with the
//          CDNA5 async global->LDS path (ASYNCcnt) to hide load latency.
// Trick: softmax normalizer is path-invariant, so the trellis uses raw logits
//        and sum_t logZ[b,t] is subtracted once at the end.

#define NEGF (-1.0e30f)

constexpr int B = 64;
constexpr int T = 2048;
constexpr int C = 96;
constexpr int S = 256;
constexpr int L = 2 * S + 1;   // 513 trellis states
constexpr int TBLK = 544;      // 17 wave32s, covers L states

static_assert(C == 96, "k_logz hardcodes 3x32 lanes per row");

__device__ __forceinline__ float lae(float a, float b) {
  // logaddexp; safe for a==b==NEGF (d==0) and mixed NEGF/normal (exp->0)
  float m = fmaxf(a, b);
  float d = fminf(a, b) - m;
  return m + __logf(1.0f + __expf(d));
}

// ---------------------------------------------------------------------------
// Kernel 1: logZ[b*T+t] = logsumexp over C=96 logits. One wave32 per row.
// ---------------------------------------------------------------------------
__global__ __launch_bounds__(256) void k_logz(const float* __restrict__ pred,
                                              float* __restrict__ logZ) {
  const int wave = threadIdx.x >> 5;
  const int lane = threadIdx.x & 31;
  const long long row = (long long)blockIdx.x * 8 + wave;   // B*T rows total
  const float* p = pred + row * C;
  float x0 = p[lane], x1 = p[lane + 32], x2 = p[lane + 64];
  float m = fmaxf(fmaxf(x0, x1), x2);
  #pragma unroll
  for (int off = 16; off; off >>= 1) m = fmaxf(m, __shfl_xor(m, off, 32));
  float s = __expf(x0 - m) + __expf(x1 - m) + __expf(x2 - m);
  #pragma unroll
  for (int off = 16; off; off >>= 1) s += __shfl_xor(s, off, 32);
  if (lane == 0) logZ[row] = m + __logf(s);
}

// ---------------------------------------------------------------------------
// Kernel 2: CTC forward trellis. One block per batch element.
// ---------------------------------------------------------------------------
__device__ __forceinline__ void async_row16(const void* gptr, void* lptr) {
  // CDNA5 async DMA: 16 bytes global -> LDS, tracked by ASYNCcnt.
  unsigned lds = (unsigned)(uintptr_t)lptr;   // low 32 bits of generic LDS ptr = LDS offset
  asm volatile("global_load_async_to_lds_b128 %0, %1, off"
               :: "v"(lds), "v"(gptr) : "memory");
}

__global__ __launch_bounds__(TBLK) void k_trellis(const float* __restrict__ pred,
                                                  const int* __restrict__ tgt,
                                                  const int* __restrict__ tmask,
                                                  const float* __restrict__ logZ,
                                                  float* __restrict__ lossB) {
  __shared__ float A[2][L + 2];     // double-buffered alpha, 2-elem NEG pad in front
  __shared__ float row[2][C];       // double-buffered logit rows
  __shared__ float red[TBLK];       // reduction scratch
  __shared__ float s_sumLogZ;
  __shared__ int s_len;

  const int tid = threadIdx.x;
  const int b = blockIdx.x;
  const float* predB = pred + (size_t)b * T * C;

  // --- target length = sum(targets_mask[b]) ---
  {
    int lm = 0;
    for (int i = tid; i < S; i += TBLK) lm += tmask[b * S + i];
    red[tid] = (float)lm;
    __syncthreads();
    if (tid < 32) {
      float s = 0.0f;
      for (int i = tid; i < TBLK; i += 32) s += red[i];
      #pragma unroll
      for (int off = 16; off; off >>= 1) s += __shfl_xor(s, off, 32);
      if (tid == 0) s_len = (int)(s + 0.5f);
    }
    __syncthreads();
    // --- sum_t logZ[b,t] (fixed order -> deterministic) ---
    float lz = 0.0f;
    for (int i = tid; i < T; i += TBLK) lz += logZ[(size_t)b * T + i];
    red[tid] = lz;
    __syncthreads();
    if (tid < 32) {
      float s = 0.0f;
      for (int i = tid; i < TBLK; i += 32) s += red[i];
      #pragma unroll
      for (int off = 16; off; off >>= 1) s += __shfl_xor(s, off, 32);
      if (tid == 0) s_sumLogZ = s;
    }
  }

  // --- per-state extended label + skip flag (registers) ---
  int myExt = 0;
  bool mySkip = false;
  if (tid < L && (tid & 1)) {
    const int j = tid >> 1;
    myExt = tgt[b * S + j];
    const int extm2 = (j >= 1) ? tgt[b * S + j - 1] : -1;
    mySkip = (myExt != 0) && (myExt != extm2);
  }

  if (tid < 2) { A[0][tid] = NEGF; A[1][tid] = NEGF; }   // NEG padding
  if (tid < C) row[0][tid] = predB[tid];                 // row 0 (sync)
  __syncthreads();

  // alpha[0]: only states 0 (blank) and 1 (first label) are live
  if (tid < L) {
    const float p = row[0][myExt];
    A[0][tid + 2] = (tid < 2) ? p : NEGF;
  }
  // prefetch row 1 asynchronously (24 lanes x 16B = 384B)
  if (tid < 24) async_row16(predB + C + tid * 4, &row[1][tid * 4]);
  asm volatile("s_wait_asynccnt 0" ::: "memory");
  __syncthreads();

  // --- 2047 recurrence steps, one barrier each ---
  for (int t = 1; t < T; ++t) {
    const int cur = t & 1, prv = cur ^ 1;
    if (t + 1 < T && tid < 24)   // prefetch row t+1 into the buffer freed at t-1
      async_row16(predB + (size_t)(t + 1) * C + tid * 4, &row[prv][tid * 4]);
    if (tid < L) {
      float a = lae(A[prv][tid + 2], A[prv][tid + 1]);
      if (mySkip) a = lae(a, A[prv][tid]);
      A[cur][tid + 2] = a + row[cur][myExt];
    }
    asm volatile("s_wait_asynccnt 0" ::: "memory");  // row t+1 landed (wave 0)
    __syncthreads();
  }

  if (tid == 0) {
    const int len = s_len;
    const int fb = (T - 1) & 1;
    const float v1 = A[fb][2 * len + 2];
    const float v2 = A[fb][2 * len + 1];
    float loss = s_sumLogZ - lae(v1, v2);      // -(alpha_true) after normalizer fixup
    if (!(isfinite(loss) && loss < 1e29f)) loss = 0.0f;   // zero_infinity
    lossB[b] = loss / (float)len;
  }
}

// ---------------------------------------------------------------------------
// Kernel 3: deterministic batch mean (no float atomics).
// ---------------------------------------------------------------------------
__global__ void k_final(const float* __restrict__ lossB, float* __restrict__ out) {
  if (threadIdx.x == 0) {
    float s = 0.0f;
    for (int i = 0; i < B; ++i) s += lossB[i];
    out[0] = s / (float)B;
  }
}

extern "C" void kernel_launch(void* const* d_in, const int* in_sizes, int n_in,
                              void* d_out, int out_size, void* d_ws, size_t ws_size,
                              hipStream_t stream) {
  const float* pred  = (const float*)d_in[0];   // [B,T,C] f32
  const int*   tgt   = (const int*)d_in[1];     // [B,S] int32
  // d_in[2] = predictions_mask: unused (pack_predictions=False -> full lengths)
  const int*   tmask = (const int*)d_in[3];     // [B,S] int32

  float* logZ  = (float*)d_ws;                  // B*T floats = 512 KB
  float* lossB = logZ + (size_t)B * T;          // B floats

  k_logz   <<<(B * T) / 8, 256, 0, stream>>>(pred, logZ);
  k_trellis<<<B, TBLK, 0, stream>>>(pred, tgt, tmask, logZ, lossB);
  k_final  <<<1, 32, 0, stream>>>(lossB, (float*)d_out);
}